// GCNEncoder_54417235640483
// MI455X (gfx1250) — compile-verified
//
#include <hip/hip_runtime.h>
#include <hip/hip_bf16.h>

typedef __attribute__((ext_vector_type(2))) float v2f;
typedef __attribute__((ext_vector_type(8))) float v8f;

#define NNODES 100000
#define NEDGES 1600000
#define FIN    58
#define FINP   60      // FIN padded to multiple of 4
#define FH     128
#define FOUT   32

// ---------------------------------------------------------------- utilities
__global__ void zero_f32(float* __restrict__ p, int n) {
  int i = blockIdx.x * blockDim.x + threadIdx.x;
  if (i < n) p[i] = 0.0f;
}

__global__ void deg_init(float* __restrict__ deg) {
  int i = blockIdx.x * blockDim.x + threadIdx.x;
  if (i < NNODES) deg[i] = 1.0f;            // self-loop contribution
}

__global__ void deg_accum(const long long* __restrict__ dst,
                          float* __restrict__ deg) {
  int e = blockIdx.x * blockDim.x + threadIdx.x;
  if (e < NEDGES) atomicAdd(&deg[(int)dst[e]], 1.0f);
}

__global__ void deg_rsqrt(float* __restrict__ deg) {
  int i = blockIdx.x * blockDim.x + threadIdx.x;
  if (i < NNODES) deg[i] = rsqrtf(deg[i]);  // deg >= 1 always (self loops)
}

// ------------------------------------------------- GEMM1: xw1[N,128] = x @ W1
// One 16-row tile per block; 8 waves cover the 128 output columns.
// fp32 WMMA 16x16x4; A tile staged in LDS (stride 60 -> conflict-free).
__global__ __launch_bounds__(256) void gemm_x_w1(const float* __restrict__ x,
                                                 const float* __restrict__ W1,
                                                 float* __restrict__ xw) {
  __shared__ float At[16 * FINP];
  const int rowBase = blockIdx.x * 16;
  const int tid = threadIdx.x;

  for (int i = tid; i < 16 * FINP; i += 256) {
    int r = i / FINP, k = i - r * FINP;
    int row = rowBase + r;
    float v = 0.0f;
    if (row < NNODES && k < FIN) v = x[row * FIN + k];
    At[i] = v;
  }
  __syncthreads();

  const int lane = tid & 31;
  const int wave = tid >> 5;
  const int half = lane >> 4;       // 0: K pair {0,1}, 1: K pair {2,3}
  const int m    = lane & 15;
  const int col  = wave * 16 + m;   // wave owns cols [16w,16w+16)

  v8f c = {};
  for (int k = 0; k < FINP; k += 4) {
    const int ka = k + 2 * half;
    v2f a;
    a.x = At[m * FINP + ka];
    a.y = At[m * FINP + ka + 1];
    v2f b;
    b.x = (ka     < FIN) ? W1[ka * FH + col]       : 0.0f;
    b.y = (ka + 1 < FIN) ? W1[(ka + 1) * FH + col] : 0.0f;
    c = __builtin_amdgcn_wmma_f32_16x16x4_f32(false, a, false, b,
                                              (short)0, c, false, false);
  }
#pragma unroll
  for (int v = 0; v < 8; ++v) {
    int row = rowBase + v + 8 * half;   // C: VGPR v -> M = v (lo half) / v+8 (hi)
    if (row < NNODES) xw[row * FH + col] = c[v];
  }
}

// -------------------------------------- scatter conv1: agg[dst] += norm*xw[src]
// One wave per edge; lane loads float4 (b128) of the 128-col row, 4 f32 atomics.
__global__ __launch_bounds__(256) void scatter1(const long long* __restrict__ src,
                                                const long long* __restrict__ dst,
                                                const float* __restrict__ dis,
                                                const float* __restrict__ xw,
                                                float* __restrict__ agg) {
  const int e = blockIdx.x * 8 + (threadIdx.x >> 5);
  if (e >= NEDGES) return;
  const int lane = threadIdx.x & 31;
  const int s = (int)src[e];
  const int d = (int)dst[e];
  const float w = dis[s] * dis[d];
  const float4 v = ((const float4*)(xw + (size_t)s * FH))[lane];
  float* ad = agg + (size_t)d * FH + lane * 4;
  atomicAdd(ad + 0, v.x * w);
  atomicAdd(ad + 1, v.y * w);
  atomicAdd(ad + 2, v.z * w);
  atomicAdd(ad + 3, v.w * w);
}

// ------------------------- epilogue1: h = relu(agg + xw*deg_is^2 + b1) in place
__global__ void relu_bias_self(const float* __restrict__ xw,
                               const float* __restrict__ b1,
                               const float* __restrict__ dis,
                               float* __restrict__ agg) {
  int t = blockIdx.x * blockDim.x + threadIdx.x;   // N*32 threads, 4 cols each
  if (t >= NNODES * 32) return;
  const int i = t >> 5;
  const int f = (t & 31) * 4;
  const float di = dis[i];
  const float d2 = di * di;
  const float4 xv = *(const float4*)(xw + (size_t)i * FH + f);
  const float4 bv = *(const float4*)(b1 + f);
  float4 av = *(float4*)(agg + (size_t)i * FH + f);
  av.x = fmaxf(av.x + xv.x * d2 + bv.x, 0.0f);
  av.y = fmaxf(av.y + xv.y * d2 + bv.y, 0.0f);
  av.z = fmaxf(av.z + xv.z * d2 + bv.z, 0.0f);
  av.w = fmaxf(av.w + xv.w * d2 + bv.w, 0.0f);
  *(float4*)(agg + (size_t)i * FH + f) = av;
}

// ------------------ GEMM2: xwmu = h@Wmu, xwstd = h@Wstd (fused, 32 rows/block)
// 8 waves: (rowtile 0/1) x (mu col0, mu col1, std col0, std col1).
// LDS stride 132 -> bank = (4m+k) mod 64, conflict-free across the 16 lanes.
__global__ __launch_bounds__(256) void gemm_h_w2(const float* __restrict__ h,
                                                 const float* __restrict__ Wmu,
                                                 const float* __restrict__ Wstd,
                                                 float* __restrict__ xwmu,
                                                 float* __restrict__ xwstd) {
  __shared__ float Ht[32 * 132];
  const int rowBase = blockIdx.x * 32;
  const int tid = threadIdx.x;

  for (int i = tid; i < 32 * FH; i += 256) {
    int r = i >> 7, k = i & 127;
    int row = rowBase + r;
    Ht[r * 132 + k] = (row < NNODES) ? h[(size_t)row * FH + k] : 0.0f;
  }
  __syncthreads();

  const int lane = tid & 31;
  const int wave = tid >> 5;
  const int half = lane >> 4;
  const int m    = lane & 15;
  const int rt   = wave >> 2;          // row tile within block
  const int sel  = wave & 3;           // 0,1 -> mu ; 2,3 -> std
  const float* __restrict__ W = (sel < 2) ? Wmu : Wstd;
  float* __restrict__ out     = (sel < 2) ? xwmu : xwstd;
  const int col = (sel & 1) * 16 + m;

  const float* Arow = &Ht[(rt * 16 + m) * 132];
  v8f c = {};
  for (int k = 0; k < FH; k += 4) {
    const int ka = k + 2 * half;
    v2f a;
    a.x = Arow[ka];
    a.y = Arow[ka + 1];
    v2f b;
    b.x = W[ka * FOUT + col];
    b.y = W[(ka + 1) * FOUT + col];
    c = __builtin_amdgcn_wmma_f32_16x16x4_f32(false, a, false, b,
                                              (short)0, c, false, false);
  }
#pragma unroll
  for (int v = 0; v < 8; ++v) {
    int row = rowBase + rt * 16 + v + 8 * half;
    if (row < NNODES) out[(size_t)row * FOUT + col] = c[v];
  }
}

// ----------------- scatter conv2/3: out{mu,std}[dst] += norm * xw{mu,std}[src]
__global__ __launch_bounds__(256) void scatter2(const long long* __restrict__ src,
                                                const long long* __restrict__ dst,
                                                const float* __restrict__ dis,
                                                const float* __restrict__ xwmu,
                                                const float* __restrict__ xwstd,
                                                float* __restrict__ outmu,
                                                float* __restrict__ outstd) {
  const int e = blockIdx.x * 8 + (threadIdx.x >> 5);
  if (e >= NEDGES) return;
  const int lane = threadIdx.x & 31;      // one column per lane (FOUT == 32)
  const int s = (int)src[e];
  const int d = (int)dst[e];
  const float w = dis[s] * dis[d];
  atomicAdd(outmu  + (size_t)d * FOUT + lane, xwmu [(size_t)s * FOUT + lane] * w);
  atomicAdd(outstd + (size_t)d * FOUT + lane, xwstd[(size_t)s * FOUT + lane] * w);
}

// ----------------------------- epilogue2: add self-loop term + bias to outputs
__global__ void epi2(const float* __restrict__ xwmu,
                     const float* __restrict__ xwstd,
                     const float* __restrict__ bmu,
                     const float* __restrict__ bstd,
                     const float* __restrict__ dis,
                     float* __restrict__ outmu,
                     float* __restrict__ outstd) {
  int t = blockIdx.x * blockDim.x + threadIdx.x;
  if (t >= NNODES * FOUT) return;
  const int i = t >> 5;
  const int f = t & 31;
  const float di = dis[i];
  const float d2 = di * di;
  outmu[t]  += xwmu[t]  * d2 + bmu[f];
  outstd[t] += xwstd[t] * d2 + bstd[f];
}

// ============================================================================
extern "C" void kernel_launch(void* const* d_in, const int* in_sizes, int n_in,
                              void* d_out, int out_size, void* d_ws, size_t ws_size,
                              hipStream_t stream) {
  (void)in_sizes; (void)n_in; (void)out_size; (void)ws_size;

  const float*     x    = (const float*)d_in[0];
  const long long* ei   = (const long long*)d_in[1];   // int64 per reference
  const long long* src  = ei;                          // edge_index[0]
  const long long* dst  = ei + NEDGES;                 // edge_index[1]
  const float*     W1   = (const float*)d_in[2];
  const float*     b1   = (const float*)d_in[3];
  const float*     Wmu  = (const float*)d_in[4];
  const float*     bmu  = (const float*)d_in[5];
  const float*     Wstd = (const float*)d_in[6];
  const float*     bstd = (const float*)d_in[7];

  float* out_mu  = (float*)d_out;
  float* out_std = out_mu + (size_t)NNODES * FOUT;

  // workspace layout (floats): dis[N] | xw1[N*128] | agg1/h[N*128]
  float* ws   = (float*)d_ws;
  float* dis  = ws;
  float* xw1  = dis + NNODES;
  float* agg1 = xw1 + (size_t)NNODES * FH;             // becomes h after relu
  // xw1 region is dead after relu_bias_self -> reuse for conv2/3 pre-agg
  float* xwmu  = xw1;
  float* xwstd = xw1 + (size_t)NNODES * FOUT;

  const int T = 256;

  // degree -> deg_is (in place in `dis`)
  deg_init <<<(NNODES + T - 1) / T, T, 0, stream>>>(dis);
  deg_accum<<<(NEDGES + T - 1) / T, T, 0, stream>>>(dst, dis);
  deg_rsqrt<<<(NNODES + T - 1) / T, T, 0, stream>>>(dis);

  // conv1
  zero_f32 <<<(NNODES * FH + T - 1) / T, T, 0, stream>>>(agg1, NNODES * FH);
  gemm_x_w1<<<(NNODES + 15) / 16, T, 0, stream>>>(x, W1, xw1);
  scatter1 <<<(NEDGES + 7) / 8, T, 0, stream>>>(src, dst, dis, xw1, agg1);
  relu_bias_self<<<(NNODES * 32 + T - 1) / T, T, 0, stream>>>(xw1, b1, dis, agg1);

  // conv2 + conv3 (mu, std)
  zero_f32 <<<(2 * NNODES * FOUT + T - 1) / T, T, 0, stream>>>(out_mu, 2 * NNODES * FOUT);
  gemm_h_w2<<<(NNODES + 31) / 32, T, 0, stream>>>(agg1, Wmu, Wstd, xwmu, xwstd);
  scatter2 <<<(NEDGES + 7) / 8, T, 0, stream>>>(src, dst, dis, xwmu, xwstd, out_mu, out_std);
  epi2     <<<(NNODES * FOUT + T - 1) / T, T, 0, stream>>>(xwmu, xwstd, bmu, bstd, dis,
                                                           out_mu, out_std);
}